// NonMaxSuppression_88210038325827
// MI455X (gfx1250) — compile-verified
//
#include <hip/hip_runtime.h>
#include <hip/hip_bf16.h>
#include <stdint.h>

// Problem geometry (fixed by the reference).
#define BB 8
#define HH 2048
#define WW 2048
#define RPB 4                               // rows per block in mask/emit passes
#define TPB 256
#define BLOCKS_PER_IMG (HH / RPB)           // 512
#define NBLK (BB * BLOCKS_PER_IMG)          // 4096
#define WORDS_PER_BLOCK (RPB * WW / 32)     // 256 (one mask word per thread)
#define MASK_WORDS (NBLK * WORDS_PER_BLOCK) // 1,048,576  (4 MiB)
#define MAX_PEAKS ((BB * HH * WW) / 8)      // 4,194,304
#define NEG_INF (-__builtin_inff())

#define AS1 __attribute__((address_space(1)))
#define AS3 __attribute__((address_space(3)))

typedef int v4i __attribute__((ext_vector_type(4)));

#if defined(__HIP_DEVICE_COMPILE__)
#if __has_builtin(__builtin_amdgcn_global_store_async_from_lds_b128) && \
    __has_builtin(__builtin_amdgcn_s_wait_asynccnt)
#define HAS_ASYNC_LDS 1
#endif
#endif
#ifndef HAS_ASYNC_LDS
#define HAS_ASYNC_LDS 0
#endif

// ---------------------------------------------------------------------------
// Kernel 0: fill d_out (2*MAX_PEAKS int32) with -1.
// gfx1250 path: stage -1s in LDS, stream out via async LDS->global stores
// (ASYNCcnt-tracked DMA-style stores; frees the wave while data drains).
// ---------------------------------------------------------------------------
#define FILL_INT4_PER_BLOCK 2048            // 32 KiB per block
#define FILL_BLOCKS ((2 * MAX_PEAKS / 4) / FILL_INT4_PER_BLOCK)  // 1024

__global__ __launch_bounds__(TPB) void nms_fill_neg1(int* __restrict__ out) {
  const int t = threadIdx.x;
  int4* o4 = (int4*)out;
  const size_t base = (size_t)blockIdx.x * FILL_INT4_PER_BLOCK;
#if HAS_ASYNC_LDS
  __shared__ int4 buf[TPB];
  buf[t] = make_int4(-1, -1, -1, -1);
  __syncthreads();
  AS3 v4i* lsrc = (AS3 v4i*)&buf[t];
#pragma unroll
  for (int k = 0; k < FILL_INT4_PER_BLOCK / TPB; ++k) {
    AS1 v4i* g = (AS1 v4i*)(o4 + base + (size_t)k * TPB + t);
    __builtin_amdgcn_global_store_async_from_lds_b128(g, lsrc, 0, 0);
  }
  __builtin_amdgcn_s_wait_asynccnt(0);
#else
  const int4 v = make_int4(-1, -1, -1, -1);
#pragma unroll
  for (int k = 0; k < FILL_INT4_PER_BLOCK / TPB; ++k) {
    o4[base + (size_t)k * TPB + t] = v;
  }
#endif
}

// ---------------------------------------------------------------------------
// Kernel 1: per-thread 32-pixel chunk -> one uint32 predicate mask word.
// Block = 4 contiguous rows of one image (flat-contiguous pixel range).
// Thread t owns pixels [t*32, t*32+32) of the block (same row, 32 cols).
// Separable 5x5 max in registers; column-max over 5 rows, then 5-tap row max.
// All global loads are 16B-aligned float4; row-edge float4s are always fully
// inside or fully outside the row (c0 is a multiple of 32), so no OOB reads.
// ---------------------------------------------------------------------------
__global__ __launch_bounds__(TPB) void nms_mask_count(
    const float* __restrict__ scores, unsigned* __restrict__ mask,
    int* __restrict__ blockCnt) {
  const int blk = blockIdx.x;
  const int t = threadIdx.x;
  const int img = blk >> 9;                 // / BLOCKS_PER_IMG
  const int r0 = (blk & (BLOCKS_PER_IMG - 1)) << 2;  // * RPB
  const int h = r0 + (t >> 6);              // thread's row
  const int c0 = (t & 63) << 5;             // thread's first column (mult of 32)

  const float* imgBase = scores + (size_t)img * HH * WW;

  // Column-max over rows h-2..h+2 for loaded cols [c0-4, c0+36).
  float cmax[40];
#pragma unroll
  for (int i = 0; i < 40; ++i) cmax[i] = NEG_INF;

#pragma unroll
  for (int dr = -2; dr <= 2; ++dr) {
    const int hh = h + dr;
    const bool rowValid = (hh >= 0) && (hh < HH);
    const float4* rp =
        (const float4*)(imgBase + (size_t)(rowValid ? hh : h) * WW);
#pragma unroll
    for (int k = 0; k < 10; ++k) {
      const int col = c0 - 4 + 4 * k;       // multiple of 4
      float4 v;
      if (rowValid && col >= 0 && col < WW) {
        v = rp[col >> 2];
      } else {
        v.x = v.y = v.z = v.w = NEG_INF;    // -inf padding (matches reference)
      }
      cmax[4 * k + 0] = fmaxf(cmax[4 * k + 0], v.x);
      cmax[4 * k + 1] = fmaxf(cmax[4 * k + 1], v.y);
      cmax[4 * k + 2] = fmaxf(cmax[4 * k + 2], v.z);
      cmax[4 * k + 3] = fmaxf(cmax[4 * k + 3], v.w);
    }
  }

  // Re-load center row (L1/L2 hit) and build the mask word.
  const float4* rp0 = (const float4*)(imgBase + (size_t)h * WW);
  unsigned m = 0;
#pragma unroll
  for (int k = 1; k <= 8; ++k) {
    const float4 v = rp0[(c0 >> 2) - 1 + k];  // loaded idx 4k..4k+3 = pixel j=4k-4..4k-1
    const float vv[4] = {v.x, v.y, v.z, v.w};
#pragma unroll
    for (int e = 0; e < 4; ++e) {
      const int j = 4 * k - 4 + e;          // pixel col = c0 + j
      const float wm =
          fmaxf(fmaxf(fmaxf(cmax[j + 2], cmax[j + 3]),
                      fmaxf(cmax[j + 4], cmax[j + 5])),
                cmax[j + 6]);
      const bool p = (vv[e] == wm) && (vv[e] > 0.0f);
      m |= (p ? 1u : 0u) << j;              // LSB = lowest flat index
    }
  }

  mask[blk * WORDS_PER_BLOCK + t] = m;

  __shared__ int scnt[TPB];
  scnt[t] = __popc(m);
  __syncthreads();
  for (int s = TPB / 2; s > 0; s >>= 1) {
    if (t < s) scnt[t] += scnt[t + s];
    __syncthreads();
  }
  if (t == 0) blockCnt[blk] = scnt[0];
}

// ---------------------------------------------------------------------------
// Kernel 2: exclusive scan of NBLK (=4096) block counts, single workgroup.
// ---------------------------------------------------------------------------
__global__ __launch_bounds__(TPB) void nms_scan_counts(
    const int* __restrict__ cnt, int* __restrict__ baseOut) {
  const int t = threadIdx.x;
  int loc[NBLK / TPB];                      // 16 per thread
  int s = 0;
#pragma unroll
  for (int i = 0; i < NBLK / TPB; ++i) {
    loc[i] = s;
    s += cnt[t * (NBLK / TPB) + i];
  }
  __shared__ int ls[TPB];
  ls[t] = s;
  __syncthreads();
  for (int off = 1; off < TPB; off <<= 1) {
    const int x = (t >= off) ? ls[t - off] : 0;
    __syncthreads();
    ls[t] += x;
    __syncthreads();
  }
  const int excl = (t == 0) ? 0 : ls[t - 1];
#pragma unroll
  for (int i = 0; i < NBLK / TPB; ++i) {
    baseOut[t * (NBLK / TPB) + i] = excl + loc[i];
  }
}

// ---------------------------------------------------------------------------
// Kernel 3: emit (h, w) at exact row-major rank; reads only the 4 MiB mask.
// d_out layout: [0, MAX_PEAKS) = h values, [MAX_PEAKS, 2*MAX_PEAKS) = w values.
// ---------------------------------------------------------------------------
__global__ __launch_bounds__(TPB) void nms_emit(
    const unsigned* __restrict__ mask, const int* __restrict__ blockBase,
    int* __restrict__ out) {
  const int blk = blockIdx.x;
  const int t = threadIdx.x;
  unsigned m = mask[blk * WORDS_PER_BLOCK + t];
  const int cnt = __popc(m);

  __shared__ int ls[TPB];
  ls[t] = cnt;
  __syncthreads();
  for (int off = 1; off < TPB; off <<= 1) {
    const int x = (t >= off) ? ls[t - off] : 0;
    __syncthreads();
    ls[t] += x;
    __syncthreads();
  }
  int base = blockBase[blk] + ls[t] - cnt;  // exclusive prefix

  const int r0 = (blk & (BLOCKS_PER_IMG - 1)) << 2;
  const int h = r0 + (t >> 6);
  const int wbase = (t & 63) << 5;
  int i = 0;
  while (m) {
    const int j = __builtin_ctz(m);
    m &= m - 1;
    const int idx = base + i;
    if (idx < MAX_PEAKS) {
      out[idx] = h;
      out[MAX_PEAKS + idx] = wbase + j;
    }
    ++i;
  }
}

// ---------------------------------------------------------------------------
// Launch. Workspace layout: [mask 4MiB][blockCnt 16KiB][blockBase 16KiB].
// ---------------------------------------------------------------------------
extern "C" void kernel_launch(void* const* d_in, const int* in_sizes, int n_in,
                              void* d_out, int out_size, void* d_ws,
                              size_t ws_size, hipStream_t stream) {
  (void)in_sizes; (void)n_in; (void)out_size; (void)ws_size;
  const float* scores = (const float*)d_in[0];
  int* out = (int*)d_out;

  unsigned* mask = (unsigned*)d_ws;
  int* blockCnt = (int*)((char*)d_ws + (size_t)MASK_WORDS * sizeof(unsigned));
  int* blockBase = blockCnt + NBLK;

  nms_fill_neg1<<<FILL_BLOCKS, TPB, 0, stream>>>(out);
  nms_mask_count<<<NBLK, TPB, 0, stream>>>(scores, mask, blockCnt);
  nms_scan_counts<<<1, TPB, 0, stream>>>(blockCnt, blockBase);
  nms_emit<<<NBLK, TPB, 0, stream>>>(mask, blockBase, out);
}